// H2GCN_43671227466239
// MI455X (gfx1250) — compile-verified
//
#include <hip/hip_runtime.h>

#define N_NODES 100000
#define IN_CH   256
#define HID     64
#define OUT_CH  40
#define E1_N    1600000
#define E2_N    3200000

typedef __attribute__((ext_vector_type(2))) float v2f;
typedef __attribute__((ext_vector_type(8))) float v8f;

// ---------------------------------------------------------------------------
// Zero workspace accumulators (h1, h2) — must run every call (atomics accumulate).
// ---------------------------------------------------------------------------
__global__ void h2gcn_zero_f32(float* __restrict__ p, size_t n) {
    size_t i = (size_t)blockIdx.x * blockDim.x + threadIdx.x;
    size_t stride = (size_t)gridDim.x * blockDim.x;
    for (; i < n; i += stride) p[i] = 0.0f;
}

// ---------------------------------------------------------------------------
// GEMM1: h0[n, j] = sum_k x[n,k] * w1[j,k]   (x:[N,256], w1:[64,256], h0:[N,64])
// One wave computes one 16x16 tile with V_WMMA_F32_16X16X4_F32, K-loop of 64.
// Block = 128 threads (4 waves) = one 16-row stripe covering all 4 HID tiles.
// A layout (ISA 7.12.2, 32-bit A 16x4): lanes 0-15 -> K=k0,k0+1 ; lanes 16-31 -> K=k0+2,k0+3.
// ---------------------------------------------------------------------------
__global__ __launch_bounds__(128)
void h2gcn_fc1_wmma(const float* __restrict__ x, const float* __restrict__ w1,
                    float* __restrict__ h0) {
    const int mtile = blockIdx.x;            // 16 rows of x / h0
    const int wave  = threadIdx.x >> 5;      // HID tile 0..3
    const int lane  = threadIdx.x & 31;
    const int mrow  = (mtile << 4) + (lane & 15);
    const int ncol  = (wave  << 4) + (lane & 15);
    const int khalf = (lane >> 4) << 1;      // 0 or 2

    const float* __restrict__ xrow = x  + (size_t)mrow * IN_CH + khalf;
    const float* __restrict__ wrow = w1 + (size_t)ncol * IN_CH + khalf;

    v8f c = {};
    #pragma unroll 8
    for (int k0 = 0; k0 < IN_CH; k0 += 4) {
        v2f a = *(const v2f*)(xrow + k0);
        v2f b = *(const v2f*)(wrow + k0);
        c = __builtin_amdgcn_wmma_f32_16x16x4_f32(false, a, false, b,
                                                  (short)0, c, false, false);
    }

    // C/D layout: VGPR i -> row (i + 8*(lane>=16)), col = lane&15 within tile
    const int rbase = (mtile << 4) + ((lane >> 4) << 3);
    const int col   = (wave  << 4) + (lane & 15);
    #pragma unroll
    for (int i = 0; i < 8; ++i)
        h0[(size_t)(rbase + i) * HID + col] = c[i];
}

// ---------------------------------------------------------------------------
// SpMM scatter: out[src[e], c] += val[e] * h0[dst[e], c]
// 64 consecutive threads (2 waves) share one edge -> coalesced 256B row gather
// from h0 (resident in 192MB L2) and hardware global_atomic_add_f32 scatter.
// ---------------------------------------------------------------------------
__global__ __launch_bounds__(256)
void h2gcn_spmm_scatter(const int* __restrict__ rows, const int* __restrict__ cols,
                        const float* __restrict__ vals, const float* __restrict__ h0,
                        float* __restrict__ out, int nedges) {
    long long tid = (long long)blockIdx.x * blockDim.x + threadIdx.x;
    int e  = (int)(tid >> 6);
    int ch = (int)(tid & 63);
    if (e >= nedges) return;
    int r = rows[e];
    int d = cols[e];
    float v = vals[e];
    unsafeAtomicAdd(&out[(size_t)r * HID + ch], v * h0[(size_t)d * HID + ch]);
}

// ---------------------------------------------------------------------------
// GEMM2: out[n, o] = sum_{j<192} h[n,j] * w2[o,j] + bias[o]
// h = concat(h0,h1,h2) -> three K=64 sub-loops over the three buffers.
// 3 waves/block = 3 col tiles (0-15, 16-31, 32-47); cols >= 40 are computed
// with a clamped B row (no divergence before WMMA) and never stored.
// ---------------------------------------------------------------------------
__global__ __launch_bounds__(96)
void h2gcn_fcout_wmma(const float* __restrict__ h0, const float* __restrict__ h1,
                      const float* __restrict__ h2, const float* __restrict__ w2,
                      const float* __restrict__ bias, float* __restrict__ out) {
    const int mtile = blockIdx.x;
    const int wave  = threadIdx.x >> 5;      // col tile 0..2
    const int lane  = threadIdx.x & 31;
    const int mrow  = (mtile << 4) + (lane & 15);
    const int col   = (wave  << 4) + (lane & 15);
    const int oc    = (col < OUT_CH) ? col : (OUT_CH - 1);   // clamp, keep EXEC full
    const int khalf = (lane >> 4) << 1;

    const float* __restrict__ ar0 = h0 + (size_t)mrow * HID + khalf;
    const float* __restrict__ ar1 = h1 + (size_t)mrow * HID + khalf;
    const float* __restrict__ ar2 = h2 + (size_t)mrow * HID + khalf;
    const float* __restrict__ br  = w2 + (size_t)oc * (3 * HID) + khalf;

    v8f c = {};
    #pragma unroll 4
    for (int k0 = 0; k0 < HID; k0 += 4) {
        v2f a = *(const v2f*)(ar0 + k0);
        v2f b = *(const v2f*)(br + k0);
        c = __builtin_amdgcn_wmma_f32_16x16x4_f32(false, a, false, b,
                                                  (short)0, c, false, false);
    }
    #pragma unroll 4
    for (int k0 = 0; k0 < HID; k0 += 4) {
        v2f a = *(const v2f*)(ar1 + k0);
        v2f b = *(const v2f*)(br + HID + k0);
        c = __builtin_amdgcn_wmma_f32_16x16x4_f32(false, a, false, b,
                                                  (short)0, c, false, false);
    }
    #pragma unroll 4
    for (int k0 = 0; k0 < HID; k0 += 4) {
        v2f a = *(const v2f*)(ar2 + k0);
        v2f b = *(const v2f*)(br + 2 * HID + k0);
        c = __builtin_amdgcn_wmma_f32_16x16x4_f32(false, a, false, b,
                                                  (short)0, c, false, false);
    }

    if (col < OUT_CH) {
        float bv = bias[col];
        const int rbase = (mtile << 4) + ((lane >> 4) << 3);
        #pragma unroll
        for (int i = 0; i < 8; ++i)
            out[(size_t)(rbase + i) * OUT_CH + col] = c[i] + bv;
    }
}

// ---------------------------------------------------------------------------
// Launcher. Workspace: h0 | h1 | h2, each N*HID floats (25.6 MB, 76.8 MB total).
// Input order: x, edge_index(unused), adj_src, adj_dst, adj_val,
//              adj2_src, adj2_dst, adj2_val, fc1_w, fc_out_w, fc_out_b
// ---------------------------------------------------------------------------
extern "C" void kernel_launch(void* const* d_in, const int* in_sizes, int n_in,
                              void* d_out, int out_size, void* d_ws, size_t ws_size,
                              hipStream_t stream) {
    const float* x   = (const float*)d_in[0];
    const int*   a1s = (const int*)  d_in[2];
    const int*   a1d = (const int*)  d_in[3];
    const float* a1v = (const float*)d_in[4];
    const int*   a2s = (const int*)  d_in[5];
    const int*   a2d = (const int*)  d_in[6];
    const float* a2v = (const float*)d_in[7];
    const float* w1  = (const float*)d_in[8];
    const float* w2  = (const float*)d_in[9];
    const float* bia = (const float*)d_in[10];
    float* out = (float*)d_out;

    float* h0 = (float*)d_ws;
    float* h1 = h0 + (size_t)N_NODES * HID;
    float* h2 = h1 + (size_t)N_NODES * HID;

    // zero atomic accumulators h1,h2 (2 * N * HID floats)
    h2gcn_zero_f32<<<2048, 256, 0, stream>>>(h1, (size_t)2 * N_NODES * HID);

    // h0 = x @ fc1_w^T
    h2gcn_fc1_wmma<<<N_NODES / 16, 128, 0, stream>>>(x, w1, h0);

    // h1 = spmm(adj, h0); h2 = spmm(adj2, h0)
    {
        int thr1 = 256;
        long long tot1 = (long long)E1_N * HID;
        long long tot2 = (long long)E2_N * HID;
        h2gcn_spmm_scatter<<<(unsigned)((tot1 + thr1 - 1) / thr1), thr1, 0, stream>>>(
            a1s, a1d, a1v, h0, h1, E1_N);
        h2gcn_spmm_scatter<<<(unsigned)((tot2 + thr1 - 1) / thr1), thr1, 0, stream>>>(
            a2s, a2d, a2v, h0, h2, E2_N);
    }

    // out = concat(h0,h1,h2) @ fc_out_w^T + b
    h2gcn_fcout_wmma<<<N_NODES / 16, 96, 0, stream>>>(h0, h1, h2, w2, bia, out);
}